// LGLoss_68719477142
// MI455X (gfx1250) — compile-verified
//
#include <hip/hip_runtime.h>
#include <math.h>

#define BB     4
#define NUM    4000
#define CDIM   128
#define NDIST  500
#define KDIST  500
#define KMAP   10
#define EPSF   1e-8f
#define HEAT_T 8.0f

typedef float v2f __attribute__((ext_vector_type(2)));
typedef float v8f __attribute__((ext_vector_type(8)));

// workspace layout in floats
#define OFF_NRM 0                          // BB*NUM = 16000
#define OFF_D2  16384                      // BB*NDIST*NUM = 8,000,000
#define OFF_P   (OFF_D2 + BB*NDIST*NUM)    // 2 branches * BB*NDIST = 4000 partials
#define OFF_Q   (OFF_P + 2*BB*NDIST)       // 128 neighbor partials
// total ~8,020,512 floats ~= 32.1 MB

// ---------- K0: per-row squared norms (one wave per row) ----------
__global__ void norms_kernel(const float* __restrict__ feat, float* __restrict__ nrm) {
  int wid  = (blockIdx.x * blockDim.x + threadIdx.x) >> 5;
  int lane = threadIdx.x & 31;
  if (wid < BB * NUM) {
    const float* p = feat + (size_t)wid * CDIM;
    float s = 0.f;
    #pragma unroll
    for (int c = lane; c < CDIM; c += 32) { float v = p[c]; s += v * v; }
    #pragma unroll
    for (int off = 16; off > 0; off >>= 1) s += __shfl_xor(s, off, 32);
    if (lane == 0) nrm[wid] = s;
  }
}

// ---------- K1: D2[b][n][m] = ||f1_n||^2 + ||f_m||^2 - 2*dot, via f32 WMMA ----------
// one wave per 16x16 tile; K looped in steps of 4 (V_WMMA_F32_16X16X4_F32)
__global__ void gemm_d2_kernel(const float* __restrict__ feat,
                               const int*   __restrict__ rn,
                               const float* __restrict__ nrm,
                               float*       __restrict__ D2) {
  const int NT_N = 32, NT_M = 250;            // 32*16=512 >= 500, 250*16 = 4000
  int w    = blockIdx.x * (blockDim.x >> 5) + (threadIdx.x >> 5);
  int lane = threadIdx.x & 31;
  int b    = w / (NT_N * NT_M);
  int rem  = w - b * (NT_N * NT_M);
  int tn   = rem / NT_M;
  int tm   = rem - tn * NT_M;
  int n0 = tn * 16, m0 = tm * 16;

  int l16 = lane & 15;
  int kh  = (lane >> 4) * 2;                  // K sub-offset 0 or 2 per lane half

  int an = n0 + l16; if (an > NDIST - 1) an = NDIST - 1;   // clamp; guarded at store
  int arow = rn[an];
  int brow = m0 + l16;

  const float* pa = feat + ((size_t)b * NUM + arow) * CDIM + kh;
  const float* pb = feat + ((size_t)b * NUM + brow) * CDIM + kh;

  v8f acc = {0.f, 0.f, 0.f, 0.f, 0.f, 0.f, 0.f, 0.f};
  #pragma unroll 4
  for (int k0 = 0; k0 < CDIM; k0 += 4) {
    v2f a  = *(const v2f*)(pa + k0);          // (K=kh, kh+1) for this lane half
    v2f bv = *(const v2f*)(pb + k0);
    acc = __builtin_amdgcn_wmma_f32_16x16x4_f32(
        /*neg_a=*/false, a, /*neg_b=*/false, bv,
        /*c_mod=*/(short)0, acc, /*reuse_a=*/false, /*reuse_b=*/false);
  }

  float nb = nrm[b * NUM + m0 + l16];         // N = lane&15 in C/D layout
  int mh = (lane >> 4) * 8;
  #pragma unroll
  for (int r = 0; r < 8; ++r) {
    int n = n0 + r + mh;                      // M = r + 8*(lane>=16)
    if (n < NDIST) {
      float na = nrm[b * NUM + rn[n]];
      float d2 = na + nb - 2.0f * acc[r];
      d2 = d2 < 0.f ? 0.f : d2;
      D2[((size_t)(b * NDIST + n)) * NUM + (m0 + l16)] = d2;
    }
  }
}

// ---------- K2: deterministic top-500 set selection + cosine terms ----------
// one 256-thread block per (b,n). Bits of nonneg f32 are order-isomorphic to u32:
// binary-search the 500th-smallest threshold T in LDS, take bits<T plus
// (500-count_less) ties chosen deterministically by lowest index.
__global__ void select_kernel(const float* __restrict__ D2,
                              const float* __restrict__ dist,
                              const int*   __restrict__ rn,
                              float*       __restrict__ P) {
  __shared__ unsigned sd[NUM];
  __shared__ int scnt;
  __shared__ float r1[256], r2[256], r3[256];
  int tid = threadIdx.x;
  int b = blockIdx.x / NDIST;
  int n = blockIdx.x - b * NDIST;

  const float* row = D2 + (size_t)(b * NDIST + n) * NUM;
  for (int i = tid; i < NUM; i += 256) sd[i] = __float_as_uint(row[i]);
  __syncthreads();

  unsigned lo = 0u, hi = 0x7F800000u;         // +inf bits: count(<=hi)=4000 >= 500
  while (lo < hi) {
    unsigned mid = lo + ((hi - lo) >> 1);
    if (tid == 0) scnt = 0;
    __syncthreads();
    int c = 0;
    for (int i = tid; i < NUM; i += 256) c += (sd[i] <= mid) ? 1 : 0;
    if (c) atomicAdd(&scnt, c);
    __syncthreads();
    int tot = scnt;
    __syncthreads();
    if (tot >= KDIST) hi = mid; else lo = mid + 1;
  }
  unsigned T = hi;

  if (tid == 0) scnt = 0;
  __syncthreads();
  int cl = 0;
  for (int i = tid; i < NUM; i += 256) cl += (sd[i] < T) ? 1 : 0;
  if (cl) atomicAdd(&scnt, cl);
  __syncthreads();
  int needTies = KDIST - scnt;

  int rncol = rn[n];
  const float* dcol = dist + (size_t)b * NUM * NUM + rncol;  // stride NUM gather
  float s1 = 0.f, s2 = 0.f, s3 = 0.f;
  for (int i = tid; i < NUM; i += 256) {
    unsigned bits = sd[i];
    bool take = bits < T;
    if (!take && bits == T) {                 // rare: rank ties by index (deterministic)
      int rk = 0;
      for (int j = 0; j < i; ++j) rk += (sd[j] == T) ? 1 : 0;
      take = rk < needTies;
    }
    if (take) {
      float d2 = __uint_as_float(bits);
      float df = dcol[(size_t)i * NUM];
      s1 += sqrtf(d2) * df;
      s2 += d2;                               // == sum(dist_result^2)
      s3 += df * df;
    }
  }
  r1[tid] = s1; r2[tid] = s2; r3[tid] = s3;
  __syncthreads();
  for (int off = 128; off > 0; off >>= 1) {
    if (tid < off) { r1[tid] += r1[tid+off]; r2[tid] += r2[tid+off]; r3[tid] += r3[tid+off]; }
    __syncthreads();
  }
  if (tid == 0) {
    float na = sqrtf(r2[0]); if (na < EPSF) na = EPSF;
    float nb = sqrtf(r3[0]); if (nb < EPSF) nb = EPSF;
    P[blockIdx.x] = 1.0f - fabsf(r1[0] / (na * nb));
  }
}

// ---------- K3: 3-D kNN(k=10, incl self) + heat-kernel neighbor loss ----------
__global__ void neighbor_kernel(const float* __restrict__ verts,
                                const float* __restrict__ tc,
                                float* __restrict__ Qout) {
  __shared__ float red[256];
  int tid = threadIdx.x;
  int idx = blockIdx.x * 256 + tid;
  float acc = 0.f;
  if (idx < BB * NUM) {
    int b = idx / NUM;
    int i = idx - b * NUM;
    const float* vb = verts + (size_t)b * NUM * 3;
    const float* tb = tc    + (size_t)b * NUM * 3;
    float px = vb[i*3+0], py = vb[i*3+1], pz = vb[i*3+2];
    float bd[KMAP]; int bi[KMAP];
    #pragma unroll
    for (int k = 0; k < KMAP; ++k) { bd[k] = 3.0e38f; bi[k] = 0; }
    for (int j = 0; j < NUM; ++j) {           // all lanes read same j: broadcast loads
      float dx = vb[j*3+0] - px;
      float dy = vb[j*3+1] - py;
      float dz = vb[j*3+2] - pz;
      float d2 = dx*dx + dy*dy + dz*dz;
      if (d2 < bd[KMAP-1]) {                  // strict <: stable w.r.t. earlier index
        bd[KMAP-1] = d2; bi[KMAP-1] = j;
        #pragma unroll
        for (int k = KMAP-1; k > 0; --k) {    // fully unrolled -> stays in VGPRs
          if (bd[k] < bd[k-1]) {
            float td = bd[k]; bd[k] = bd[k-1]; bd[k-1] = td;
            int ti = bi[k]; bi[k] = bi[k-1]; bi[k-1] = ti;
          }
        }
      }
    }
    // drop bd[0] (nearest == self), use neighbors 1..9 as a set
    float tx = tb[i*3+0], ty = tb[i*3+1], tz = tb[i*3+2];
    #pragma unroll
    for (int k = 1; k < KMAP; ++k) {
      int j = bi[k];
      float dx = tb[j*3+0] - tx;
      float dy = tb[j*3+1] - ty;
      float dz = tb[j*3+2] - tz;
      float tsq = dx*dx + dy*dy + dz*dz;
      acc += __expf(-bd[k] * (1.0f / HEAT_T)) * tsq;
    }
  }
  red[tid] = acc;
  __syncthreads();
  for (int off = 128; off > 0; off >>= 1) {
    if (tid < off) red[tid] += red[tid + off];
    __syncthreads();
  }
  if (tid == 0) Qout[blockIdx.x] = red[0];
}

__global__ void zeroQ_kernel(float* __restrict__ Q) {
  if (threadIdx.x < 128) Q[threadIdx.x] = 0.f;
}

// ---------- final deterministic combine ----------
__global__ void finalize_kernel(const float* __restrict__ P,
                                const float* __restrict__ Q,
                                float* __restrict__ out) {
  __shared__ float red[256];
  int tid = threadIdx.x;
  float s = 0.f;
  for (int i = tid; i < 2 * BB * NDIST; i += 256) s += P[i];
  float q = 0.f;
  if (tid < 128) q = Q[tid];
  s += q * (1.0f / (float)(BB * NUM * (KMAP - 1)));   // mean over 4*4000*9 per call
  red[tid] = s;
  __syncthreads();
  for (int off = 128; off > 0; off >>= 1) {
    if (tid < off) red[tid] += red[tid + off];
    __syncthreads();
  }
  if (tid == 0) out[0] = red[0];
}

extern "C" void kernel_launch(void* const* d_in, const int* in_sizes, int n_in,
                              void* d_out, int out_size, void* d_ws, size_t ws_size,
                              hipStream_t stream) {
  // setup_inputs order: cfeats1, cfeats2, feat1, feat2, dist1, dist2,
  // verts1, verts2, verts1_corr, verts2_corr, verts1_corr_v1, verts2_corr_v2, rn1, rn2
  const float* feat1 = (const float*)d_in[2];
  const float* feat2 = (const float*)d_in[3];
  const float* dist1 = (const float*)d_in[4];
  const float* dist2 = (const float*)d_in[5];
  const float* verts1 = (const float*)d_in[6];
  const float* verts2 = (const float*)d_in[7];
  const float* verts1_corr = (const float*)d_in[8];
  const float* verts2_corr = (const float*)d_in[9];
  const int* rn1 = (const int*)d_in[12];
  const int* rn2 = (const int*)d_in[13];

  float* ws  = (float*)d_ws;   // needs ~32.1 MB
  float* nrm = ws + OFF_NRM;
  float* D2  = ws + OFF_D2;
  float* P   = ws + OFF_P;
  float* Q   = ws + OFF_Q;

  zeroQ_kernel<<<1, 128, 0, stream>>>(Q);

  const float* feats[2] = {feat1, feat2};
  const float* dists[2] = {dist1, dist2};
  const int*   rns[2]   = {rn1, rn2};
  for (int br = 0; br < 2; ++br) {
    norms_kernel<<<(BB * NUM) / 4, 128, 0, stream>>>(feats[br], nrm);
    // 4 batches * 32 n-tiles * 250 m-tiles = 32000 waves / 8 waves per block
    gemm_d2_kernel<<<4000, 256, 0, stream>>>(feats[br], rns[br], nrm, D2);
    select_kernel<<<BB * NDIST, 256, 0, stream>>>(D2, dists[br], rns[br],
                                                  P + br * (BB * NDIST));
  }
  neighbor_kernel<<<63, 256, 0, stream>>>(verts1, verts2_corr, Q + 0);
  neighbor_kernel<<<63, 256, 0, stream>>>(verts2, verts1_corr, Q + 64);
  finalize_kernel<<<1, 256, 0, stream>>>(P, Q, (float*)d_out);
}